// VectorQuantizer_29429115912768
// MI455X (gfx1250) — compile-verified
//
#include <hip/hip_runtime.h>
#include <math.h>

typedef __attribute__((ext_vector_type(2))) float v2f;
typedef __attribute__((ext_vector_type(8))) float v8f;

#define ED   256
#define NE   1024
#define BATCH 32
#define HW0  256           // 16x16
#define HW1  1024          // 32x32
#define N0   (BATCH*HW0)   // 8192
#define N1   (BATCH*HW1)   // 32768

// ---------------- output layout (floats) ----------------
#define ZQ0   ((long long)BATCH*ED*HW0)          // 2,097,152
#define ZQ1   ((long long)BATCH*ED*HW1)          // 8,388,608
#define ENC0  ((long long)N0*NE)                 // 8,388,608
#define ENC1  ((long long)N1*NE)                 // 33,554,432
#define OFF_LOSS 0LL
#define OFF_ZQ0  1LL
#define OFF_ZQ1  (OFF_ZQ0 + ZQ0)
#define OFF_PERP (OFF_ZQ1 + ZQ1)
#define OFF_ENC0 (OFF_PERP + 1)
#define OFF_ENC1 (OFF_ENC0 + ENC0)
#define OFF_IDX0 (OFF_ENC1 + ENC1)
#define OFF_IDX1 (OFF_IDX0 + N0)

// ---------------- workspace layout (bytes) ----------------
#define WS_ZF0   0ULL
#define WS_ZF1   (WS_ZF0 + (size_t)N0*ED*4)      // +8 MB
#define WS_CBN0  (WS_ZF1 + (size_t)N1*ED*4)      // +32 MB
#define WS_CBN1  (WS_CBN0 + 4096)
#define WS_HIST0 (WS_CBN1 + 4096)
#define WS_HIST1 (WS_HIST0 + 4096)
#define WS_LACC  (WS_HIST1 + 4096)               // 2 floats
#define WS_IDX0  (WS_LACC + 4096)
#define WS_IDX1  (WS_IDX0 + (size_t)N0*4)

// ============================================================
// 1) [B,C,HW] -> [B*HW, C] transpose (LDS tiled, conflict-free)
// ============================================================
__global__ __launch_bounds__(256) void transpose_kernel(
    const float* __restrict__ in, float* __restrict__ out, int HW)
{
    __shared__ float tile[32][33];
    const int b   = blockIdx.z;
    const int hw0 = blockIdx.x * 32;
    const int c0  = blockIdx.y * 32;
    const float* inb  = in  + (size_t)b * ED * HW;
    float*       outb = out + (size_t)b * HW * ED;
    const int tx = threadIdx.x, ty = threadIdx.y;   // 32 x 8
    #pragma unroll
    for (int i = 0; i < 32; i += 8)
        tile[ty + i][tx] = inb[(size_t)(c0 + ty + i) * HW + hw0 + tx];
    __syncthreads();
    #pragma unroll
    for (int i = 0; i < 32; i += 8)
        outb[(size_t)(hw0 + ty + i) * ED + c0 + tx] = tile[tx][ty + i];
}

// ============================================================
// 2) ||codebook row||^2 for both codebooks (2048 rows)
// ============================================================
__global__ __launch_bounds__(256) void cbnorm_kernel(
    const float* __restrict__ cb0, const float* __restrict__ cb1,
    float* __restrict__ n0, float* __restrict__ n1)
{
    const int r = blockIdx.x * 256 + threadIdx.x;   // 0..2047
    const float* cb  = (r < NE) ? cb0 : cb1;
    float*       dst = (r < NE) ? n0  : n1;
    const int rr = r & (NE - 1);
    const float4* p = (const float4*)(cb + (size_t)rr * ED);
    float s = 0.f;
    #pragma unroll 8
    for (int i = 0; i < ED/4; ++i) {
        float4 v = p[i];
        s += v.x*v.x + v.y*v.y + v.z*v.z + v.w*v.w;
    }
    dst[rr] = s;
}

// ============================================================
// Async global -> LDS staging of one 16x256 fp32 codebook tile.
// CDNA5 path: global_load_async_to_lds_b128 (ASYNCcnt-tracked, no VGPR
// round trip). ISA 10.2: for LDS-aperture flat addresses the hardware LDS
// byte address is addr[31:0], so truncating a generic pointer to __shared__
// memory gives the VDST LDS address operand directly.
// ============================================================
__device__ __forceinline__ void stage_tile_async(
    const float* __restrict__ cb, int tile, float* buf, int tid)
{
    const float4* g = (const float4*)(cb + (size_t)tile * 16 * ED);
    const unsigned lbase = (unsigned)(unsigned long long)buf;
    #pragma unroll
    for (int j = 0; j < 4; ++j) {
        const int e = tid + j * 256;                 // float4 index 0..1023
        const unsigned laddr = lbase + (unsigned)e * 16u;
        const float4* gp = g + e;
        asm volatile("global_load_async_to_lds_b128 %0, %1, off"
                     :: "v"(laddr), "v"(gp) : "memory");
    }
}

__device__ __forceinline__ void wait_async_all()
{
    asm volatile("s_wait_asynccnt 0x0" ::: "memory");
}

// ============================================================
// 3) WMMA argmin kernel: fp32 GEMM zf @ cb^T via v_wmma_f32_16x16x4_f32
//    8 waves/block; each wave owns 16 rows. Codebook tiles double-buffered
//    in LDS via async-to-LDS copies overlapped with the WMMA chain.
// ============================================================
__global__ __launch_bounds__(256) void vq_argmin_wmma(
    const float* __restrict__ zf,    // [N][256]
    const float* __restrict__ cb,    // [1024][256]
    const float* __restrict__ cbn,   // [1024]
    float* __restrict__ idx_f,       // [N]  (d_out slice)
    int*   __restrict__ idx_i,       // [N]  (workspace)
    unsigned* __restrict__ hist,     // [1024]
    float* __restrict__ lossAcc)     // [1]
{
    __shared__ float lds_cb[2][16 * ED];             // 2 x 16 KB
    const int tid  = threadIdx.x;
    const int lane = tid & 31;
    const int wave = tid >> 5;
    const int nloc = lane & 15;                      // M for A / N for B,C,D
    const int kb   = (lane >> 4) << 1;               // K sub-offset per half-wave
    const int rowBase = (blockIdx.x * 8 + wave) * 16;

    // ---- load A strip into registers in WMMA-A (f32 16x16x4) layout ----
    const float* zrow = zf + (size_t)(rowBase + nloc) * ED;
    v2f a[64];
    float z2p = 0.f;
    #pragma unroll
    for (int s = 0; s < 64; ++s) {
        v2f av = *(const v2f*)(zrow + 4*s + kb);
        a[s] = av;
        z2p += av[0]*av[0] + av[1]*av[1];
    }
    // each half-wave holds half the K values of its row -> combine
    const float z2 = z2p + __shfl_xor(z2p, 16, 32);  // ||z_row||^2, row = nloc

    float best[8];
    int   bidx[8];
    #pragma unroll
    for (int r = 0; r < 8; ++r) { best[r] = 3.4e38f; bidx[r] = 0; }

    // prologue: stage tile 0
    stage_tile_async(cb, 0, lds_cb[0], tid);
    wait_async_all();
    __syncthreads();

    for (int tile = 0; tile < NE/16; ++tile) {
        // kick off async copy of the next tile into the other buffer;
        // its latency hides behind the 64-deep WMMA chain below.
        if (tile + 1 < NE/16)
            stage_tile_async(cb, tile + 1, lds_cb[(tile + 1) & 1], tid);

        const float* bbuf = lds_cb[tile & 1];
        const float cn = cbn[tile*16 + nloc];

        // two independent accumulator chains for XDL ILP
        v8f acc0 = {0.f,0.f,0.f,0.f,0.f,0.f,0.f,0.f};
        v8f acc1 = {0.f,0.f,0.f,0.f,0.f,0.f,0.f,0.f};
        #pragma unroll
        for (int s = 0; s < 64; s += 2) {
            v2f b0 = *(const v2f*)(bbuf + nloc*ED + 4*s       + kb);
            v2f b1 = *(const v2f*)(bbuf + nloc*ED + 4*(s + 1) + kb);
            acc0 = __builtin_amdgcn_wmma_f32_16x16x4_f32(
                       false, a[s],     false, b0, (short)0, acc0, false, false);
            acc1 = __builtin_amdgcn_wmma_f32_16x16x4_f32(
                       false, a[s + 1], false, b1, (short)0, acc1, false, false);
        }
        #pragma unroll
        for (int r = 0; r < 8; ++r) {
            const float score = cn - 2.0f * (acc0[r] + acc1[r]); // ||c||^2 - 2 z.c
            if (score < best[r]) { best[r] = score; bidx[r] = tile*16 + nloc; }
        }

        // next tile's async copy done + visible to all waves before reuse
        wait_async_all();
        __syncthreads();
    }

    // ---- reduce (min,idx) across the 16 lanes that see each row ----
    #pragma unroll
    for (int m = 1; m < 16; m <<= 1) {
        #pragma unroll
        for (int r = 0; r < 8; ++r) {
            const float ob = __shfl_xor(best[r], m, 32);
            const int   oi = __shfl_xor(bidx[r], m, 32);
            if (ob < best[r] || (ob == best[r] && oi < bidx[r])) {
                best[r] = ob; bidx[r] = oi;
            }
        }
    }

    // writer lanes: 0..7 -> rows M=0..7 (r=lane), 16..23 -> rows M=8..15
    const bool hi  = lane >= 16;
    const int  r_w = lane & 7;
    const int  M_w = r_w + (hi ? 8 : 0);
    const int  srcLane = (M_w & 15) | (lane & 16);   // lane holding z2 of row M_w
    const float z2w = __shfl(z2, srcLane, 32);
    if ((!hi && lane < 8) || (hi && lane < 24)) {
        const int row = rowBase + M_w;
        const int bi  = bidx[r_w];
        idx_f[row] = (float)bi;
        idx_i[row] = bi;
        atomicAdd(&hist[bi], 1u);
        atomicAdd(lossAcc, z2w + best[r_w]);         // ||z - c_best||^2
    }
}

// ============================================================
// 4) one-hot fill (coalesced float4 stores; covers the 0xAA poison)
// ============================================================
__global__ __launch_bounds__(256) void onehot_kernel(
    const int* __restrict__ idxv, float* __restrict__ enc, int Nrows)
{
    const long long t = (long long)blockIdx.x * 256 + threadIdx.x; // N*256 threads
    if (t >= (long long)Nrows * (NE/4)) return;
    const int n  = (int)(t >> 8);          // t / 256
    const int c0 = ((int)t & 255) << 2;    // 4-float chunk
    float4 v = make_float4(0.f, 0.f, 0.f, 0.f);
    const int d = idxv[n] - c0;
    if (d >= 0 && d < 4) ((float*)&v)[d] = 1.0f;
    ((float4*)enc)[t] = v;
}

// ============================================================
// 5) z_q gather + transpose back to [B,C,H,W] (coalesced stores)
// ============================================================
__global__ __launch_bounds__(256) void scatter_zq(
    const float* __restrict__ cb, const int* __restrict__ idxv,
    float* __restrict__ zq, int HW, long long total)
{
    const long long o = (long long)blockIdx.x * 256 + threadIdx.x;
    if (o >= total) return;
    const int hw = (int)(o % HW);
    const int c  = (int)((o / HW) % ED);
    const int b  = (int)(o / ((long long)HW * ED));
    const int n  = b * HW + hw;
    zq[o] = cb[(size_t)idxv[n] * ED + c];
}

// ============================================================
// 6) finalize: loss + perplexity scalars
// ============================================================
__global__ __launch_bounds__(1024) void finalize_kernel(
    const unsigned* __restrict__ h0, const unsigned* __restrict__ h1,
    const float* __restrict__ lacc, float* __restrict__ out_loss,
    float* __restrict__ out_perp)
{
    __shared__ float s0[1024];
    __shared__ float s1[1024];
    const int t = threadIdx.x;
    const float p0 = (float)h0[t] / (float)N0;
    const float p1 = (float)h1[t] / (float)N1;
    s0[t] = -p0 * logf(p0 + 1e-10f);
    s1[t] = -p1 * logf(p1 + 1e-10f);
    __syncthreads();
    for (int st = 512; st > 0; st >>= 1) {
        if (t < st) { s0[t] += s0[t + st]; s1[t] += s1[t + st]; }
        __syncthreads();
    }
    if (t == 0) {
        *out_perp = 0.5f * (expf(s0[0]) + expf(s1[0]));
        *out_loss = 1.25f * (lacc[0] / (float)((long long)N0 * ED) +
                             lacc[1] / (float)((long long)N1 * ED));
    }
}

// ============================================================
extern "C" void kernel_launch(void* const* d_in, const int* in_sizes, int n_in,
                              void* d_out, int out_size, void* d_ws, size_t ws_size,
                              hipStream_t stream)
{
    (void)in_sizes; (void)n_in; (void)out_size; (void)ws_size;
    const float* z0  = (const float*)d_in[0];
    const float* z1  = (const float*)d_in[1];
    const float* cb0 = (const float*)d_in[2];   // emb_z_w  (used for z0)
    const float* cb1 = (const float*)d_in[3];   // emb_w    (used for z1)
    float* out = (float*)d_out;
    char*  ws  = (char*)d_ws;

    float*    zf0   = (float*)(ws + WS_ZF0);
    float*    zf1   = (float*)(ws + WS_ZF1);
    float*    cbn0  = (float*)(ws + WS_CBN0);
    float*    cbn1  = (float*)(ws + WS_CBN1);
    unsigned* hist0 = (unsigned*)(ws + WS_HIST0);
    unsigned* hist1 = (unsigned*)(ws + WS_HIST1);
    float*    lacc  = (float*)(ws + WS_LACC);
    int*      idx0  = (int*)(ws + WS_IDX0);
    int*      idx1  = (int*)(ws + WS_IDX1);

    // zero histograms + loss accumulators (contiguous region)
    hipMemsetAsync(ws + WS_HIST0, 0, 4096 + 4096 + 8, stream);

    // transpose to [N, 256]
    transpose_kernel<<<dim3(HW0/32, ED/32, BATCH), dim3(32, 8), 0, stream>>>(z0, zf0, HW0);
    transpose_kernel<<<dim3(HW1/32, ED/32, BATCH), dim3(32, 8), 0, stream>>>(z1, zf1, HW1);

    // codebook row norms
    cbnorm_kernel<<<8, 256, 0, stream>>>(cb0, cb1, cbn0, cbn1);

    // WMMA distance + argmin (8 waves of 16 rows per block)
    vq_argmin_wmma<<<N0/128, 256, 0, stream>>>(zf0, cb0, cbn0,
        out + OFF_IDX0, idx0, hist0, &lacc[0]);
    vq_argmin_wmma<<<N1/128, 256, 0, stream>>>(zf1, cb1, cbn1,
        out + OFF_IDX1, idx1, hist1, &lacc[1]);

    // one-hot encodings
    onehot_kernel<<<(int)(((long long)N0*(NE/4) + 255)/256), 256, 0, stream>>>(
        idx0, out + OFF_ENC0, N0);
    onehot_kernel<<<(int)(((long long)N1*(NE/4) + 255)/256), 256, 0, stream>>>(
        idx1, out + OFF_ENC1, N1);

    // z_q gather back to [B, C, H, W]
    scatter_zq<<<(int)((ZQ0 + 255)/256), 256, 0, stream>>>(cb0, idx0, out + OFF_ZQ0, HW0, ZQ0);
    scatter_zq<<<(int)((ZQ1 + 255)/256), 256, 0, stream>>>(cb1, idx1, out + OFF_ZQ1, HW1, ZQ1);

    // scalars
    finalize_kernel<<<1, 1024, 0, stream>>>(hist0, hist1, lacc,
        out + OFF_LOSS, out + OFF_PERP);
}